// PagedAttention_35639638622523
// MI455X (gfx1250) — compile-verified
//
#include <hip/hip_runtime.h>

#define KVH  8
#define QPK  4            // query heads per kv head
#define HS   128
#define BS   16
#define MBPS 128
#define QSCALE 0.08838834764831845f

typedef float v2f __attribute__((ext_vector_type(2)));
typedef float v4f __attribute__((ext_vector_type(4)));
typedef float v8f __attribute__((ext_vector_type(8)));

// Swap values between lane L and L^16 (wave32 half exchange) via ds_swizzle.
__device__ __forceinline__ float swz_xor16(float x) {
  int i = __builtin_bit_cast(int, x);
  i = __builtin_amdgcn_ds_swizzle(i, 0x401F);  // group-of-32, xor_mask=0x10, and_mask=0x1f
  return __builtin_bit_cast(float, i);
}

__device__ __forceinline__ float ntf(const float* p) {
  return __builtin_nontemporal_load(p);
}
__device__ __forceinline__ v2f ntf2(const float* p) {
  return __builtin_nontemporal_load((const v2f*)p);
}

__global__ __launch_bounds__(256) void paged_attn_wmma_f32(
    const float* __restrict__ q,      // [B, 32*128]
    const float* __restrict__ knew,   // [B, 8*128]
    const float* __restrict__ vnew,   // [B, 8*128]
    const float* __restrict__ kcache, // [NB, 8, 128, 16]
    const float* __restrict__ vcache, // [NB, 8, 128, 16]
    const int*   __restrict__ btab,   // [B, 128]
    const int*   __restrict__ ctxlen, // [B]
    float* __restrict__ out)          // [B, 32*128]
{
  const int h    = blockIdx.x;
  const int b    = blockIdx.y;
  const int tid  = threadIdx.x;
  const int wave = __builtin_amdgcn_readfirstlane(tid >> 5);  // wave-uniform in SGPR
  const int lane = tid & 31;
  const int ln   = lane & 15;   // token row (QK-A) / head col (B) / d-row mod 16 (PV-A)
  const int half = lane >> 4;

  __shared__ float qs[16 * HS];            // Q^T tile, rows 4..15 zero, pre-scaled
  __shared__ float accbuf[8][QPK][HS];     // per-wave partial outputs
  __shared__ float mstat[8][QPK];
  __shared__ float lstat[8][QPK];

  for (int i = tid; i < 16 * HS; i += 256) qs[i] = 0.0f;
  __syncthreads();
  for (int i = tid; i < QPK * HS; i += 256) {
    int n = i >> 7, d = i & (HS - 1);
    qs[n * HS + d] = q[((size_t)b * 32 + h * QPK + n) * HS + d] * QSCALE;
  }
  __syncthreads();

  const int L       = ctxlen[b];
  const int nblk    = (L + BS - 1) / BS;
  const int lastblk = (L - 1) >> 4;
  const int lastoff = (L - 1) & 15;
  const float* kn = knew + ((size_t)b * KVH + h) * HS;
  const float* vn = vnew + ((size_t)b * KVH + h) * HS;

  // The only v_new values this lane can ever need in the PV chain (d = dt*16+ln).
  float vnw[8];
  #pragma unroll
  for (int dt = 0; dt < 8; ++dt) vnw[dt] = vn[dt * BS + ln];

  // Per-lane LDS base for the Q^T B-operand: bq(kc) = qs[ln*HS + half*2 + kc*4 .. +1]
  const float* qsl = qs + ln * HS + half * 2;

  float m_run = -1e30f, l_run = 0.0f;
  v8f acc[8];
  #pragma unroll
  for (int dt = 0; dt < 8; ++dt)
    #pragma unroll
    for (int e = 0; e < 8; ++e) acc[dt][e] = 0.0f;

  for (int j = wave; j < nblk; j += 8) {
    const int phys  = __builtin_amdgcn_readfirstlane(btab[b * MBPS + j]);
    const float* kb = kcache + ((size_t)phys * KVH + h) * (HS * BS);
    const float* vb = vcache + ((size_t)phys * KVH + h) * (HS * BS);
    if (j + 8 < nblk) {  // prefetch this wave's next K/V block
      const int pn = __builtin_amdgcn_readfirstlane(btab[b * MBPS + j + 8]);
      __builtin_prefetch(kcache + ((size_t)pn * KVH + h) * (HS * BS), 0, 2);
      __builtin_prefetch(vcache + ((size_t)pn * KVH + h) * (HS * BS), 0, 2);
    }
    const bool isLast = (j == lastblk);

    // Per-lane strided bases -> every cache load is saddr + one voff + const imm.
    const float* kbl = kb + half * 32 + ln;       // K: a(kc) = kbl[kc*64], kbl[kc*64+16]
    const float* vbl = vb + ln * BS + half * 2;   // V: av(dt,tc) = vbl[dt*256 + tc*4 ..+1]

    // ---- scores^T(16tok x 16col) = K(16x128) * Q^T(128x16): 32 chained k=4 WMMAs
    v8f S = {0.f, 0.f, 0.f, 0.f, 0.f, 0.f, 0.f, 0.f};
    #pragma unroll
    for (int kc = 0; kc < 32; ++kc) {
      v2f a;
      a.x = ntf(kbl + kc * 64);
      a.y = ntf(kbl + kc * 64 + 16);
      const v2f bq = *(const v2f*)(qsl + kc * 4);
      S = __builtin_amdgcn_wmma_f32_16x16x4_f32(false, a, false, bq, (short)0, S,
                                                false, false);
    }

    // New-token K override: recompute that one score as q[head] . k_new (q pre-scaled).
    if (isLast) {
      float dot = 0.0f;
      const float* qrow = qs + ln * HS;
      #pragma unroll
      for (int d = 0; d < HS; d += 4) {
        const v4f qv = *(const v4f*)(qrow + d);
        dot += qv.x * kn[d] + qv.y * kn[d + 1] + qv.z * kn[d + 2] + qv.w * kn[d + 3];
      }
      const int mh = lastoff >> 3, mm = lastoff & 7;
      #pragma unroll
      for (int m = 0; m < 8; ++m)
        if (m == mm && half == mh) S[m] = dot;
    }

    // mask invalid tokens (token index in block = m + 8*half)
    const int tbase = j * BS + half * 8;
    #pragma unroll
    for (int m = 0; m < 8; ++m)
      if (tbase + m >= L) S[m] = -1e30f;

    // online softmax (per head column, duplicated across lane halves)
    float bm = S[0];
    #pragma unroll
    for (int m = 1; m < 8; ++m) bm = fmaxf(bm, S[m]);
    bm = fmaxf(bm, swz_xor16(bm));
    const float m_new   = fmaxf(m_run, bm);
    const float rescale = __expf(m_run - m_new);

    float P[8];
    float lb = 0.0f;
    #pragma unroll
    for (int m = 0; m < 8; ++m) { P[m] = __expf(S[m] - m_new); lb += P[m]; }
    lb += swz_xor16(lb);
    l_run = l_run * rescale + lb;
    m_run = m_new;

    #pragma unroll
    for (int dt = 0; dt < 8; ++dt)
      #pragma unroll
      for (int e = 0; e < 8; ++e) acc[dt][e] *= rescale;

    // Per-lane V-override masks (token pair containing the new token).
    bool f0[4], f1[4];
    #pragma unroll
    for (int tc = 0; tc < 4; ++tc) {
      const int t0 = tc * 4 + half * 2;
      f0[tc] = isLast && (t0 == lastoff);
      f1[tc] = isLast && (t0 + 1 == lastoff);
    }

    // ---- out^T(128 x 16col) += V^T(128 x 16tok) * probs^T(16tok x 16col)
    #pragma unroll
    for (int tc = 0; tc < 4; ++tc) {
      // B operand: lane needs probs[token tc*4 + half*2 + {0,1}][head ln]
      v2f bp;
      {
        const int ta0 = tc * 4, tb0 = tc * 4 + 2;
        float va = P[ta0 & 7]; if ((ta0 >> 3) == 1) va = swz_xor16(va);
        float vb0 = P[tb0 & 7]; if ((tb0 >> 3) == 0) vb0 = swz_xor16(vb0);
        bp.x = half ? vb0 : va;
        const int ta1 = tc * 4 + 1, tb1 = tc * 4 + 3;
        float vc1 = P[ta1 & 7]; if ((ta1 >> 3) == 1) vc1 = swz_xor16(vc1);
        float vd1 = P[tb1 & 7]; if ((tb1 >> 3) == 0) vd1 = swz_xor16(vd1);
        bp.y = half ? vd1 : vc1;
      }
      #pragma unroll
      for (int dt = 0; dt < 8; ++dt) {
        v2f av = ntf2(vbl + dt * 256 + tc * 4);   // two consecutive tokens of row d
        av.x = f0[tc] ? vnw[dt] : av.x;           // v_cndmask, no branch
        av.y = f1[tc] ? vnw[dt] : av.y;
        acc[dt] = __builtin_amdgcn_wmma_f32_16x16x4_f32(false, av, false, bp,
                                                        (short)0, acc[dt],
                                                        false, false);
      }
    }
  }

  // ---- spill per-wave partials, flash-decoding combine across the 8 waves
  if (half == 0 && ln < QPK) { mstat[wave][ln] = m_run; lstat[wave][ln] = l_run; }
  if (ln < QPK) {
    #pragma unroll
    for (int dt = 0; dt < 8; ++dt)
      #pragma unroll
      for (int m = 0; m < 8; ++m)
        accbuf[wave][ln][dt * 16 + m + 8 * half] = acc[dt][m];
  }
  __syncthreads();

  for (int o = tid; o < QPK * HS; o += 256) {
    const int n = o >> 7, d = o & (HS - 1);
    float mg = mstat[0][n];
    #pragma unroll
    for (int w = 1; w < 8; ++w) mg = fmaxf(mg, mstat[w][n]);
    float num = 0.0f, den = 0.0f;
    #pragma unroll
    for (int w = 0; w < 8; ++w) {
      const float f = __expf(mstat[w][n] - mg);
      num += f * accbuf[w][n][d];
      den += f * lstat[w][n];
    }
    out[((size_t)b * 32 + h * QPK + n) * HS + d] = num / den;
  }
}

extern "C" void kernel_launch(void* const* d_in, const int* in_sizes, int n_in,
                              void* d_out, int out_size, void* d_ws, size_t ws_size,
                              hipStream_t stream) {
  const float* q  = (const float*)d_in[0];
  const float* k  = (const float*)d_in[1];
  const float* v  = (const float*)d_in[2];
  const float* kc = (const float*)d_in[3];
  const float* vc = (const float*)d_in[4];
  const int*   bt = (const int*)d_in[5];
  const int*   cl = (const int*)d_in[6];
  float* out = (float*)d_out;

  const int batch = in_sizes[6];  // context_lens count == B
  dim3 grid(KVH, batch);
  paged_attn_wmma_f32<<<grid, 256, 0, stream>>>(q, k, v, kc, vc, bt, cl, out);
}